// LIS_67912022885155
// MI455X (gfx1250) — compile-verified
//
#include <hip/hip_runtime.h>

#define NUM_IN   8192
#define NUM_OUT  8192
#define SPLITK   8
#define KCHUNK   (NUM_IN / SPLITK)   // 1024

typedef __attribute__((ext_vector_type(2))) float v2f;
typedef __attribute__((ext_vector_type(8))) float v8f;

// ---------------------------------------------------------------------------
// Kernel 1: elementwise LIS trace update
//   vec[i] = (pre_trace[i]*sigmoid(d[i]) + in_spikes[i]) * (1 - sigmoid(d[i]))
// ---------------------------------------------------------------------------
__global__ void __launch_bounds__(256)
lis_prep(const float* __restrict__ in_spikes,
         const float* __restrict__ pre_trace,
         const float* __restrict__ pre_decay_raw,
         float* __restrict__ vec) {
    int i = blockIdx.x * blockDim.x + threadIdx.x;
    if (i < NUM_IN) {
        float d  = 1.0f / (1.0f + __expf(-pre_decay_raw[i]));
        float nt = fmaf(pre_trace[i], d, in_spikes[i]);
        vec[i]   = nt * (1.0f - d);
    }
}

// ---------------------------------------------------------------------------
// Kernel 2: mat-vec via V_WMMA_F32_16X16X4_F32, split-K = 8.
//   Block = 8 waves. blockIdx.x = row-group (16 rows). wave-in-block = K chunk.
//   A operand: lane L(0-15) -> row r0+L, cols {k,k+1}; lane L+16 -> {k+2,k+3}
//   (exactly the ISA 16x4 f32 A layout) -> one b64 load per lane per WMMA.
//   B operand: vector broadcast across all 16 columns (same K mapping as A).
//   D column N=0 lives in lanes 0 (M=0..7) and 16 (M=8..15).
// ---------------------------------------------------------------------------
__global__ void __launch_bounds__(256)
lis_matvec_wmma(const float* __restrict__ W,
                const float* __restrict__ vec,
                float* __restrict__ partial) {
    const int lane   = threadIdx.x & 31;
    const int kchunk = threadIdx.x >> 5;      // 0..7 (wave in block)
    const int rowg   = blockIdx.x;            // 0..511
    const int r0     = rowg * 16;
    const int kbase  = kchunk * KCHUNK;

    const int row    = r0 + (lane & 15);
    const int klane  = (lane >> 4) << 1;      // 0 for lanes 0-15, 2 for 16-31

    const float* wp = W   + (size_t)row * NUM_IN + kbase + klane;
    const float* vp = vec + kbase + klane;

    v8f acc = {};
    #pragma unroll 8
    for (int k = 0; k < KCHUNK; k += 4) {
        // Weights are streamed once (256MB > 192MB L2): non-temporal.
        v2f a = __builtin_nontemporal_load((const v2f*)(wp + k));
        v2f b = *(const v2f*)(vp + k);
        acc = __builtin_amdgcn_wmma_f32_16x16x4_f32(
            /*neg_a=*/false, a, /*neg_b=*/false, b,
            /*c_mod=*/(short)0, acc, /*reuse_a=*/false, /*reuse_b=*/false);
    }

    // Column N=0 of D: lane 0 holds M=0..7, lane 16 holds M=8..15.
    if ((lane & 15) == 0) {
        float* p = partial + (size_t)kchunk * NUM_OUT + r0 + ((lane >> 4) << 3);
        #pragma unroll
        for (int i = 0; i < 8; ++i) p[i] = acc[i];
    }
}

// ---------------------------------------------------------------------------
// Kernel 3: reduce split-K partials + bias, then softmax over 8192 logits.
// Single block of 1024 threads, 8 elements per thread.
// ---------------------------------------------------------------------------
__global__ void __launch_bounds__(1024)
lis_softmax(const float* __restrict__ partial,
            const float* __restrict__ bias,
            float* __restrict__ out) {
    __shared__ float sred[32];
    __shared__ float maxv, sumv;

    const int tid = threadIdx.x;

    float logit[8];
    #pragma unroll
    for (int j = 0; j < 8; ++j) {
        int o = tid + j * 1024;
        float s = bias[o];
        #pragma unroll
        for (int c = 0; c < SPLITK; ++c) s += partial[c * NUM_OUT + o];
        logit[j] = s;
    }

    // --- max reduction ---
    float m = logit[0];
    #pragma unroll
    for (int j = 1; j < 8; ++j) m = fmaxf(m, logit[j]);
    #pragma unroll
    for (int off = 16; off > 0; off >>= 1)
        m = fmaxf(m, __shfl_xor(m, off, 32));
    if ((tid & 31) == 0) sred[tid >> 5] = m;
    __syncthreads();
    if (tid < 32) {
        float t = sred[tid];
        #pragma unroll
        for (int off = 16; off > 0; off >>= 1)
            t = fmaxf(t, __shfl_xor(t, off, 32));
        if (tid == 0) maxv = t;
    }
    __syncthreads();
    m = maxv;

    // --- exp + sum reduction ---
    float e[8];
    float s = 0.0f;
    #pragma unroll
    for (int j = 0; j < 8; ++j) { e[j] = __expf(logit[j] - m); s += e[j]; }
    #pragma unroll
    for (int off = 16; off > 0; off >>= 1)
        s += __shfl_xor(s, off, 32);
    if ((tid & 31) == 0) sred[tid >> 5] = s;   // safe: prior reads were pre-barrier
    __syncthreads();
    if (tid < 32) {
        float t = sred[tid];
        #pragma unroll
        for (int off = 16; off > 0; off >>= 1)
            t += __shfl_xor(t, off, 32);
        if (tid == 0) sumv = t;
    }
    __syncthreads();

    float inv = 1.0f / sumv;
    #pragma unroll
    for (int j = 0; j < 8; ++j) out[tid + j * 1024] = e[j] * inv;
}

// ---------------------------------------------------------------------------
extern "C" void kernel_launch(void* const* d_in, const int* in_sizes, int n_in,
                              void* d_out, int out_size, void* d_ws, size_t ws_size,
                              hipStream_t stream) {
    const float* in_spikes     = (const float*)d_in[0];
    const float* pre_trace     = (const float*)d_in[1];
    const float* weight        = (const float*)d_in[2];
    const float* bias          = (const float*)d_in[3];
    const float* pre_decay_raw = (const float*)d_in[4];
    float* out = (float*)d_out;

    // workspace layout: [0,8192) vec ; [8192, 8192 + 8*8192) split-K partials
    float* vec     = (float*)d_ws;
    float* partial = vec + NUM_IN;

    lis_prep<<<NUM_IN / 256, 256, 0, stream>>>(in_spikes, pre_trace,
                                               pre_decay_raw, vec);
    lis_matvec_wmma<<<NUM_OUT / 16, 256, 0, stream>>>(weight, vec, partial);
    lis_softmax<<<1, 1024, 0, stream>>>(partial, bias, out);
}